// ODEFunc_532575944735
// MI455X (gfx1250) — compile-verified
//
#include <hip/hip_runtime.h>
#include <hip/hip_bf16.h>
#include <stdint.h>

#define GN 50000
#define GD 128
#define GE 1600000
#define PIPE 4   // async pipeline depth per wave

// ---------------------------------------------------------------------------
// CDNA5 async global->LDS copy (ASYNCcnt-tracked), inline asm for portability.
// Per-lane: copies 16 bytes from the global address gaddr to the LDS offset lds_off.
// ---------------------------------------------------------------------------
__device__ __forceinline__ unsigned lds_off_of(const void* p) {
    return (unsigned)(uintptr_t)(__attribute__((address_space(3))) const void*)p;
}

__device__ __forceinline__ void async_load_b128_to_lds(const float* gsrc, float* ldst) {
    unsigned lds_off = lds_off_of(ldst);
    unsigned long long gaddr = (unsigned long long)(uintptr_t)gsrc;
    asm volatile("global_load_async_to_lds_b128 %0, %1, off"
                 :
                 : "v"(lds_off), "v"(gaddr)
                 : "memory");
}

#if defined(__has_builtin) && __has_builtin(__builtin_amdgcn_s_wait_asynccnt)
#define WAIT_ASYNC(n) __builtin_amdgcn_s_wait_asynccnt(n)
#else
#define WAIT_ASYNC(n) asm volatile("s_wait_asynccnt %0" ::"i"(n) : "memory")
#endif

__device__ __forceinline__ void wait_async_le(int n) {  // n is wave-uniform
    switch (n) {
        case 0: WAIT_ASYNC(0); break;
        case 1: WAIT_ASYNC(1); break;
        case 2: WAIT_ASYNC(2); break;
        default: WAIT_ASYNC(3); break;
    }
}

__device__ __forceinline__ void wait_ds_0() {
#if defined(__has_builtin) && __has_builtin(__builtin_amdgcn_s_wait_dscnt)
    __builtin_amdgcn_s_wait_dscnt(0);
#else
    asm volatile("s_wait_dscnt 0x0" ::: "memory");
#endif
}

// ---------------------------------------------------------------------------
// Gather-multiply-scatter. One wave per edge per iteration; lane l owns
// features [4l, 4l+4). PIPE async gathers of h[src] rows kept in flight per
// wave; atomics (global_atomic_add_f32, no-return -> STOREcnt, never waited)
// scatter into the L2-resident 25.6 MB accumulator (= d_out[0, N*D)).
// ---------------------------------------------------------------------------
__global__ __launch_bounds__(256) void scatter_kernel(const float* __restrict__ h,
                                                      const float* __restrict__ e,
                                                      const int* __restrict__ src,
                                                      const int* __restrict__ dst,
                                                      float* __restrict__ acc) {
    __shared__ alignas(16) float buf[8][PIPE][GD];  // 16 KB per workgroup
    const int w = threadIdx.x >> 5;
    const int lane = threadIdx.x & 31;
    const int stride = gridDim.x * 8;

    const int edge0 = blockIdx.x * 8 + w;  // wave-uniform
    if (edge0 >= GE) return;

    // prime: keep up to PIPE gathers in flight
    int outstanding = 0;
    for (int k = 0; k < PIPE; ++k) {
        int eIdx = edge0 + k * stride;
        if (eIdx < GE) {
            int s = src[eIdx];
            async_load_b128_to_lds(h + (size_t)s * GD + lane * 4, &buf[w][k][lane * 4]);
            ++outstanding;
        }
    }

    int slot = 0;
    for (int edge = edge0; edge < GE; edge += stride) {
        // oldest in-flight load is this slot's: wait until it has landed in LDS
        wait_async_le(outstanding - 1);
        --outstanding;

        const float4 hv = *(const float4*)&buf[w][slot][lane * 4];

        // refill this slot for edge + PIPE*stride; ds_load above must have
        // completed into VGPRs before the async write can touch the slot.
        const int nx = edge + PIPE * stride;
        if (nx < GE) {
            wait_ds_0();
            int s = src[nx];
            async_load_b128_to_lds(h + (size_t)s * GD + lane * 4, &buf[w][slot][lane * 4]);
            ++outstanding;
        }

        const float ev = e[edge];
        const int d = dst[edge];
        float* o = acc + (size_t)d * GD + lane * 4;
        unsafeAtomicAdd(o + 0, hv.x * ev);
        unsafeAtomicAdd(o + 1, hv.y * ev);
        unsafeAtomicAdd(o + 2, hv.z * ev);
        unsafeAtomicAdd(o + 3, hv.w * ev);

        slot = (slot + 1) & (PIPE - 1);
    }
}

// ---------------------------------------------------------------------------
// Zero the whole output (accumulator region + E-element zero tail).
// ---------------------------------------------------------------------------
__global__ __launch_bounds__(256) void init_kernel(float4* __restrict__ out4, int n4) {
    int i = blockIdx.x * blockDim.x + threadIdx.x;
    int stride = gridDim.x * blockDim.x;
    float4 z = make_float4(0.f, 0.f, 0.f, 0.f);
    for (; i < n4; i += stride) out4[i] = z;
}

// ---------------------------------------------------------------------------
// In-place gated residual: out[i] = sigmoid(alpha) * (out[i] - h[i]).
// ---------------------------------------------------------------------------
__global__ __launch_bounds__(256) void finalize_kernel(float4* __restrict__ out4,
                                                       const float4* __restrict__ h4,
                                                       const float* __restrict__ alpha,
                                                       int n4) {
    const float a = alpha[0];
    const float g = 1.0f / (1.0f + __expf(-a));
    int i = blockIdx.x * blockDim.x + threadIdx.x;
    int stride = gridDim.x * blockDim.x;
    for (; i < n4; i += stride) {
        float4 o = out4[i];
        float4 hh = h4[i];
        o.x = g * (o.x - hh.x);
        o.y = g * (o.y - hh.y);
        o.z = g * (o.z - hh.z);
        o.w = g * (o.w - hh.w);
        out4[i] = o;
    }
}

extern "C" void kernel_launch(void* const* d_in, const int* in_sizes, int n_in,
                              void* d_out, int out_size, void* d_ws, size_t ws_size,
                              hipStream_t stream) {
    // inputs (setup_inputs order): t, x, alpha, src, dst
    const float* x     = (const float*)d_in[1];
    const float* alpha = (const float*)d_in[2];
    const int*   src   = (const int*)d_in[3];
    const int*   dst   = (const int*)d_in[4];
    float* out = (float*)d_out;

    const float* h = x;                    // [N, D]
    const float* e = x + (size_t)GN * GD;  // [E]

    // 1) zero entire output (accumulator + zero tail): 2,000,000 float4
    const int total4 = (GN * GD + GE) / 4;
    init_kernel<<<(total4 + 255) / 256, 256, 0, stream>>>((float4*)out, total4);

    // 2) gather-multiply-scatter with 4-deep async LDS pipeline; 65536 waves
    scatter_kernel<<<8192, 256, 0, stream>>>(h, e, src, dst, out);

    // 3) gated residual over h region: 1,600,000 float4
    const int h4 = (GN * GD) / 4;
    finalize_kernel<<<(h4 + 255) / 256, 256, 0, stream>>>((float4*)out, (const float4*)h,
                                                          alpha, h4);
}